// DeformableConv1d_90537910600074
// MI455X (gfx1250) — compile-verified
//
#include <hip/hip_runtime.h>
#include <hip/hip_bf16.h>

typedef __attribute__((ext_vector_type(2))) float v2f;
typedef __attribute__((ext_vector_type(8))) float v8f;

namespace {
constexpr int kB      = 32;
constexpr int kC      = 64;       // Cin == Cout == offset channels
constexpr int kL      = 16384;
constexpr int kLog2L  = 14;
constexpr int kTile   = 32;       // positions per wave-tile (2 x WMMA-N)
constexpr int kWaves  = 2;        // waves per workgroup
constexpr int kThreads = kWaves * 32;
constexpr int kIters  = 16;       // tiles per wave
constexpr int kTilesTotal = kB * kL / kTile;             // 16384
constexpr int kTilesPerWG = kWaves * kIters;             // 32
constexpr int kWGs        = kTilesTotal / kTilesPerWG;   // 512
}

__device__ __forceinline__ v8f wmma_f32(v2f a, v2f b, v8f c) {
  return __builtin_amdgcn_wmma_f32_16x16x4_f32(
      /*neg_a=*/false, a, /*neg_b=*/false, b,
      /*c_mod=*/(short)0, c, /*reuse_a=*/false, /*reuse_b=*/false);
}

// D[64co x 32l] = W(64x64) @ Xtile(64x32), two 16-wide N-subtiles sharing every
// A-fragment. A-frags: LDS weights [co][ci] row-major (contig k-pair -> b64).
// B-frags: per-wave LDS tile [pos][ci] (contig k-pair -> b64).
__device__ __forceinline__ void gemm64x64x32(const float* __restrict__ sW,
                                             const float* __restrict__ sX,
                                             int n, int kb,
                                             v8f acc0[4], v8f acc1[4]) {
  for (int s = 0; s < 16; ++s) {
    const int k0 = s * 4 + kb;
    const v2f b0 = *(const v2f*)(sX + n * kC + k0);
    const v2f b1 = *(const v2f*)(sX + (16 + n) * kC + k0);
#pragma unroll
    for (int t = 0; t < 4; ++t) {
      const v2f af = *(const v2f*)(sW + (t * 16 + n) * kC + k0);
      acc0[t] = wmma_f32(af, b0, acc0[t]);
      acc1[t] = wmma_f32(af, b1, acc1[t]);
    }
  }
}

__global__ __launch_bounds__(kThreads)
void deformable_conv1d_wmma(const float* __restrict__ x,
                            const float* __restrict__ wOff,
                            const float* __restrict__ bOff,
                            const float* __restrict__ wReg,
                            const float* __restrict__ bReg,
                            float* __restrict__ out) {
  __shared__ float sWo[kC * kC];                // 16 KB
  __shared__ float sWr[kC * kC];                // 16 KB
  __shared__ float sBo[kC];
  __shared__ float sBr[kC];
  __shared__ float sTile[kWaves][kTile * kC];   // [pos][ci], 8 KB per wave

  const int tid = threadIdx.x;
  for (int i = tid; i < kC * kC; i += kThreads) {
    sWo[i] = wOff[i];   // [Cout][Cin][1] row-major == [co][ci]
    sWr[i] = wReg[i];
  }
  if (tid < kC) { sBo[tid] = bOff[tid]; sBr[tid] = bReg[tid]; }
  __syncthreads();

  const int lane = tid & 31;
  const int wave = tid >> 5;
  const int n    = lane & 15;   // WMMA N / A-row index
  const int half = lane >> 4;   // 0: K=0,1   1: K=2,3
  const int kb   = half * 2;
  float* xdw = sTile[wave];

  const v8f vzero = {0.f, 0.f, 0.f, 0.f, 0.f, 0.f, 0.f, 0.f};

  for (int it = 0; it < kIters; ++it) {
    const int tile = (blockIdx.x * kIters + it) * kWaves + wave;
    const long long p0 = (long long)tile * kTile;   // flat (b,l) position
    const int b  = (int)(p0 >> kLog2L);             // / 16384
    const int l0 = (int)(p0 & (kL - 1));
    // per-batch slab bases; all further addressing is 32-bit lane offsets
    // against these SGPR-uniform bases -> GVS-mode global_load/store
    const float* __restrict__ xb   = x   + ((long long)b << kLog2L) * kC;
    float* __restrict__       outb = out + ((long long)b << kLog2L) * kC;

    // ---- stage raw X tile [64ci x 32l] into LDS as [pos][ci] (b128 loads) ----
#pragma unroll
    for (int j = 0; j < 16; ++j) {
      const int ci = j * 4 + (lane >> 3);
      const int c0 = (lane & 7) * 4;
      const int go = (ci << kLog2L) + l0 + c0;      // 32-bit element offset
      const float4 v = *(const float4*)(xb + go);
      xdw[(c0 + 0) * kC + ci] = v.x;
      xdw[(c0 + 1) * kC + ci] = v.y;
      xdw[(c0 + 2) * kC + ci] = v.z;
      xdw[(c0 + 3) * kC + ci] = v.w;
      if (it + 1 < kIters)  // speculative prefetch of this wave's next tile
        __builtin_prefetch(xb + go + kWaves * kTile, 0, 1);
    }

    // ---- GEMM 1: offsets = Wo @ Xtile ----
    v8f acc0[4], acc1[4];
#pragma unroll
    for (int t = 0; t < 4; ++t) { acc0[t] = vzero; acc1[t] = vzero; }
    gemm64x64x32(sWo, xdw, n, kb, acc0, acc1);

    // ---- deform: bias, clip, gather floor/ceil, lerp; rewrite tile in-place ----
    // C-layout: accU[t][r] at this lane = offsets[c = 16t + r + 8*half][pos = 16u + n]
#pragma unroll
    for (int u = 0; u < 2; ++u) {
      v8f* acc = u ? acc1 : acc0;
#pragma unroll
      for (int t = 0; t < 4; ++t) {
#pragma unroll
        for (int r = 0; r < 8; ++r) {
          const int c = t * 16 + r + half * 8;
          const int l = u * 16 + n;
          const float off = acc[t][r] + sBo[c];
          float pos = (float)(l0 + l) + off;
          pos = fminf(fmaxf(pos, 0.0f), (float)(kL - 1));
          const float fl = floorf(pos);
          const float fr = pos - fl;
          const int rowBase = c << kLog2L;          // 32-bit row offset
          const float v0 = xb[rowBase + (int)fl];
          const float v1 = xb[rowBase + (int)ceilf(pos)];
          xdw[l * kC + c] = (1.0f - fr) * v0 + fr * v1;
        }
      }
    }

    // ---- GEMM 2: out = Wr @ Xdeform ----
#pragma unroll
    for (int t = 0; t < 4; ++t) { acc0[t] = vzero; acc1[t] = vzero; }
    gemm64x64x32(sWr, xdw, n, kb, acc0, acc1);

    // ---- bias + non-temporal store (stream-out, never re-read) ----
#pragma unroll
    for (int u = 0; u < 2; ++u) {
      const v8f* acc = u ? acc1 : acc0;
#pragma unroll
      for (int t = 0; t < 4; ++t) {
#pragma unroll
        for (int r = 0; r < 8; ++r) {
          const int c = t * 16 + r + half * 8;
          const int go = (c << kLog2L) + l0 + u * 16 + n;
          __builtin_nontemporal_store(acc[t][r] + sBr[c], outb + go);
        }
      }
    }
  }
}

extern "C" void kernel_launch(void* const* d_in, const int* in_sizes, int n_in,
                              void* d_out, int out_size, void* d_ws, size_t ws_size,
                              hipStream_t stream) {
  const float* x  = (const float*)d_in[0];
  const float* wo = (const float*)d_in[1];
  const float* bo = (const float*)d_in[2];
  const float* wr = (const float*)d_in[3];
  const float* br = (const float*)d_in[4];
  float* out = (float*)d_out;
  deformable_conv1d_wmma<<<kWGs, kThreads, 0, stream>>>(x, wo, bo, wr, br, out);
}